// SAGE_81716047773789
// MI455X (gfx1250) — compile-verified
//
#include <hip/hip_runtime.h>
#include <hip/hip_bf16.h>

// ---------------------------------------------------------------------------
// GraphSAGE 3-layer forward for MI455X (gfx1250, wave32, WMMA).
//   layer: h = mean_agg(x)[dst] @ Wl + b + x @ Wr ; BN ; ReLU (x2) ; log_softmax
// ---------------------------------------------------------------------------

typedef float v2f __attribute__((ext_vector_type(2)));
typedef float v8f __attribute__((ext_vector_type(8)));

#define N_NODES 50000
#define D_H     128
#define N_EDGES 800000
#define BN_EPS  1e-5f

#define MT_PER_BLOCK 5                 // 5 x 16 = 80 rows per block
#define ROWS_PB      (MT_PER_BLOCK * 16)
#define LDS_STRIDE   130               // 128 + 2 pad: conflict-free, 8B aligned

static __device__ __forceinline__ void atomic_fadd(float* p, float v) {
    __hip_atomic_fetch_add(p, v, __ATOMIC_RELAXED, __HIP_MEMORY_SCOPE_AGENT);
}

// ------------------------------- zero fill ---------------------------------
__global__ void k_zero(float* __restrict__ p, int n) {
    int i = blockIdx.x * blockDim.x + threadIdx.x;
    int stride = gridDim.x * blockDim.x;
    for (; i < n; i += stride) p[i] = 0.0f;
}

// --------------------------- edge scatter (mean num) -----------------------
// One wave per edge. Lane l moves floats [4l, 4l+3] of the 128-wide feature.
__global__ void k_scatter(const int* __restrict__ ei,      // [2, E] (src row 0)
                          const float* __restrict__ X,     // [N, 128]
                          float* __restrict__ agg,         // [N, 128]
                          float* __restrict__ cnt,         // [N]
                          int nEdges, int addCount) {
    int wave = threadIdx.x >> 5;
    int lane = threadIdx.x & 31;
    int e = blockIdx.x * (blockDim.x >> 5) + wave;
    if (e >= nEdges) return;
    int s = ei[e];
    int d = ei[nEdges + e];
    const float4 v = ((const float4*)(X + (long)s * D_H))[lane];
    float* a = agg + (long)d * D_H + lane * 4;
    atomic_fadd(a + 0, v.x);
    atomic_fadd(a + 1, v.y);
    atomic_fadd(a + 2, v.z);
    atomic_fadd(a + 3, v.w);
    if (addCount && lane == 0) atomic_fadd(cnt + d, 1.0f);
}

// ----------------------- fused SAGE GEMM (f32 WMMA) ------------------------
// Out[80 rows x NOUT] = [mean | X] (80x256) @ [Wl ; Wr] (256 x NOUT) + bias.
// blockDim.x = 32 * (NOUT/16). Each wave owns one 16-wide column tile and
// 5 row tiles (5 x v8f accumulators). K runs in two 128-deep halves (Wl then
// Wr) so the A panel is staged in 80x128 LDS phases. NOUT is a compile-time
// constant so all B-load / store offsets are immediates off invariant bases.
template <int NOUT>
__global__ void k_sage_gemm(const float* __restrict__ agg,
                            const float* __restrict__ cnt,
                            const float* __restrict__ X,
                            const float* __restrict__ Wl,   // [128, NOUT]
                            const float* __restrict__ Wr,   // [128, NOUT]
                            const float* __restrict__ bias, // [NOUT]
                            float* __restrict__ Out) {      // [N, NOUT]
    __shared__ float As[ROWS_PB * LDS_STRIDE];
    __shared__ float rcp[ROWS_PB];

    const int blockRow = blockIdx.x * ROWS_PB;   // 625 blocks * 80 = 50000
    const int tid  = threadIdx.x;
    const int wave = tid >> 5;
    const int lane = tid & 31;
    const int l15  = lane & 15;                  // A row in tile == B column
    const int khalf = (lane >> 4) * 2;           // 0 lanes 0-15, 2 lanes 16-31
    const int col = wave * 16 + l15;             // output column (< NOUT)

    if (tid < ROWS_PB) {
        float c = cnt[blockRow + tid];
        rcp[tid] = 1.0f / fmaxf(c, 1.0f);
    }

    v8f acc[MT_PER_BLOCK] = {};

    #pragma unroll
    for (int h = 0; h < 2; ++h) {
        __syncthreads();                         // prior phase (or rcp) ready
        // Stage A panel phase: h==0 -> mean = agg * rcp ; h==1 -> self X.
        for (int i = tid; i < ROWS_PB * D_H; i += blockDim.x) {
            int row = i >> 7;
            int k   = i & (D_H - 1);
            long gr = (long)(blockRow + row);
            float v = (h == 0) ? agg[gr * D_H + k] * rcp[row]
                               : X[gr * D_H + k];
            As[row * LDS_STRIDE + k] = v;
        }
        __syncthreads();

        const float* Wp = ((h == 0) ? Wl : Wr) + khalf * NOUT + col;

        #pragma unroll 4
        for (int kb = 0; kb < D_H; kb += 4) {
            v2f b;
            b.x = Wp[kb * NOUT];                 // row kb   + khalf (imm off)
            b.y = Wp[(kb + 1) * NOUT];           // row kb+1 + khalf (imm off)
            #pragma unroll
            for (int t = 0; t < MT_PER_BLOCK; ++t) {
                int abase = (t * 16 + l15) * LDS_STRIDE + kb + khalf;
                v2f a;
                a.x = As[abase];
                a.y = As[abase + 1];
                acc[t] = __builtin_amdgcn_wmma_f32_16x16x4_f32(
                    false, a, false, b, (short)0, acc[t], false, false);
            }
        }
    }

    const float bv = bias[col];
    const int hi8 = (lane >> 4) << 3;            // +8 rows for lanes 16-31
    #pragma unroll
    for (int t = 0; t < MT_PER_BLOCK; ++t) {
        float* Orow = Out + (long)(blockRow + t * 16 + hi8) * NOUT + col;
        #pragma unroll
        for (int i = 0; i < 8; ++i) {
            Orow[i * NOUT] = acc[t][i] + bv;     // immediate offsets
        }
    }
}

// ------------------------------ BN statistics ------------------------------
__global__ void k_bn_stats(const float* __restrict__ H,
                           float* __restrict__ sum, float* __restrict__ sq,
                           int nRows) {
    int col = threadIdx.x;                   // 128 threads
    int r0 = blockIdx.x * 128;
    int r1 = r0 + 128; if (r1 > nRows) r1 = nRows;
    float s = 0.0f, q = 0.0f;
    for (int r = r0; r < r1; ++r) {
        float v = H[(long)r * D_H + col];
        s += v; q += v * v;
    }
    atomic_fadd(&sum[col], s);
    atomic_fadd(&sq[col], q);
}

__global__ void k_bn_finalize(const float* __restrict__ sum,
                              const float* __restrict__ sq,
                              const float* __restrict__ g,
                              const float* __restrict__ be,
                              float* __restrict__ scale,
                              float* __restrict__ shift) {
    int c = threadIdx.x;
    float inv = 1.0f / (float)N_NODES;
    float mu = sum[c] * inv;
    float var = sq[c] * inv - mu * mu;
    float sc = g[c] * rsqrtf(var + BN_EPS);
    scale[c] = sc;
    shift[c] = be[c] - mu * sc;
}

__global__ void k_bn_relu(float* __restrict__ H,
                          const float* __restrict__ scale,
                          const float* __restrict__ shift, int n) {
    int i = blockIdx.x * blockDim.x + threadIdx.x;
    int stride = gridDim.x * blockDim.x;
    for (; i < n; i += stride) {
        int c = i & (D_H - 1);
        H[i] = fmaxf(H[i] * scale[c] + shift[c], 0.0f);
    }
}

// ------------------------------ log_softmax --------------------------------
// One wave per row of 64; lane holds elements l and l+32.
__global__ void k_log_softmax(float* __restrict__ O, int nRows) {
    int wave = threadIdx.x >> 5;
    int lane = threadIdx.x & 31;
    int row = blockIdx.x * (blockDim.x >> 5) + wave;
    if (row >= nRows) return;
    float v0 = O[(long)row * 64 + lane];
    float v1 = O[(long)row * 64 + 32 + lane];
    float m = fmaxf(v0, v1);
    #pragma unroll
    for (int off = 16; off > 0; off >>= 1)
        m = fmaxf(m, __shfl_xor(m, off, 32));
    float s = __expf(v0 - m) + __expf(v1 - m);
    #pragma unroll
    for (int off = 16; off > 0; off >>= 1)
        s += __shfl_xor(s, off, 32);
    float L = m + __logf(s);
    O[(long)row * 64 + lane]      = v0 - L;
    O[(long)row * 64 + 32 + lane] = v1 - L;
}

// ------------------------------- launcher ----------------------------------
extern "C" void kernel_launch(void* const* d_in, const int* in_sizes, int n_in,
                              void* d_out, int out_size, void* d_ws, size_t ws_size,
                              hipStream_t stream) {
    const float* x   = (const float*)d_in[0];
    const int*   ei  = (const int*)d_in[1];
    const float* Wl0 = (const float*)d_in[2];
    const float* Wr0 = (const float*)d_in[3];
    const float* b0  = (const float*)d_in[4];
    const float* g0  = (const float*)d_in[5];
    const float* be0 = (const float*)d_in[6];
    const float* Wl1 = (const float*)d_in[7];
    const float* Wr1 = (const float*)d_in[8];
    const float* b1  = (const float*)d_in[9];
    const float* g1  = (const float*)d_in[10];
    const float* be1 = (const float*)d_in[11];
    const float* Wl2 = (const float*)d_in[12];
    const float* Wr2 = (const float*)d_in[13];
    const float* b2  = (const float*)d_in[14];
    float* out = (float*)d_out;

    // Workspace carve-up (floats): agg | H | cnt | sum | sq | scale | shift
    float* agg   = (float*)d_ws;
    float* H     = agg + (long)N_NODES * D_H;
    float* cnt   = H + (long)N_NODES * D_H;
    float* bsum  = cnt + N_NODES;
    float* bsq   = bsum + D_H;
    float* scale = bsq + D_H;
    float* shift = scale + D_H;

    const int ZB = 1024;                           // zero-fill blocks
    const int scatterBlocks = (N_EDGES + 7) / 8;   // 8 waves/block
    const int statsBlocks = (N_NODES + 127) / 128;
    const int GB = N_NODES / ROWS_PB;              // 625 (exact)
    const int EW = N_NODES * D_H;                  // 6.4M elems

    // ---------------- Layer 0 ----------------
    k_zero<<<ZB, 256, 0, stream>>>(agg, EW);
    k_zero<<<64, 256, 0, stream>>>(cnt, N_NODES);
    k_scatter<<<scatterBlocks, 256, 0, stream>>>(ei, x, agg, cnt, N_EDGES, 1);
    k_sage_gemm<128><<<GB, 256, 0, stream>>>(agg, cnt, x, Wl0, Wr0, b0, H);
    k_zero<<<1, 256, 0, stream>>>(bsum, 2 * D_H);  // bsum+bsq contiguous
    k_bn_stats<<<statsBlocks, 128, 0, stream>>>(H, bsum, bsq, N_NODES);
    k_bn_finalize<<<1, 128, 0, stream>>>(bsum, bsq, g0, be0, scale, shift);
    k_bn_relu<<<ZB, 256, 0, stream>>>(H, scale, shift, EW);

    // ---------------- Layer 1 ----------------
    k_zero<<<ZB, 256, 0, stream>>>(agg, EW);
    k_scatter<<<scatterBlocks, 256, 0, stream>>>(ei, H, agg, cnt, N_EDGES, 0);
    k_sage_gemm<128><<<GB, 256, 0, stream>>>(agg, cnt, H, Wl1, Wr1, b1, H);
    k_zero<<<1, 256, 0, stream>>>(bsum, 2 * D_H);
    k_bn_stats<<<statsBlocks, 128, 0, stream>>>(H, bsum, bsq, N_NODES);
    k_bn_finalize<<<1, 128, 0, stream>>>(bsum, bsq, g1, be1, scale, shift);
    k_bn_relu<<<ZB, 256, 0, stream>>>(H, scale, shift, EW);

    // ---------------- Layer 2 (out=64) + log_softmax ----------------
    k_zero<<<ZB, 256, 0, stream>>>(agg, EW);
    k_scatter<<<scatterBlocks, 256, 0, stream>>>(ei, H, agg, cnt, N_EDGES, 0);
    k_sage_gemm<64><<<GB, 128, 0, stream>>>(agg, cnt, H, Wl2, Wr2, b2, out);
    k_log_softmax<<<(N_NODES + 7) / 8, 256, 0, stream>>>(out, N_NODES);
}